// EEM_3573412790698
// MI455X (gfx1250) — compile-verified
//
#include <hip/hip_runtime.h>
#include <math.h>

typedef __attribute__((ext_vector_type(2))) float v2f;
typedef __attribute__((ext_vector_type(8))) float v8f;

namespace {
constexpr int kNA    = 128;   // atoms per molecule
constexpr int kNAug  = 129;   // NA + 1 (Lagrange row/col)
constexpr int kS     = 132;   // LDS row stride in floats (cols 0..128 = matrix, 129 = rhs)
constexpr int kThreads = 256; // 8 wave32
// LDS layout (float offsets)
constexpr int kOffA   = 0;                 // 129*132 = 17028
constexpr int kOffQ   = 17028;             // 132  (x solution; q = x[0..127])
constexpr int kOffY   = 17160;             // 128  (y = K q)
constexpr int kOffAlf = 17288;             // 128
constexpr int kOffJa  = 17416;             // 128
constexpr int kOffXi  = 17544;             // 128
constexpr int kOffRx  = 17672;             // 128
constexpr int kOffRy  = 17800;             // 128
constexpr int kOffRz  = 17928;             // 128
constexpr int kSmemFloats = 18056;
}

__global__ __launch_bounds__(kThreads)
void eem_mol_kernel(const int* __restrict__ Z,
                    const float* __restrict__ R,
                    const float* __restrict__ Xia,
                    const float* __restrict__ Qtot,
                    const float* __restrict__ alpha,
                    const float* __restrict__ Jp,
                    float* __restrict__ out, int T)
{
    extern __shared__ float smem[];
    float* sA   = smem + kOffA;
    float* sQ   = smem + kOffQ;
    float* sY   = smem + kOffY;
    float* sAlf = smem + kOffAlf;
    float* sJa  = smem + kOffJa;
    float* sXi  = smem + kOffXi;
    float* sRx  = smem + kOffRx;
    float* sRy  = smem + kOffRy;
    float* sRz  = smem + kOffRz;

    const int mol  = blockIdx.x;
    const int tid  = threadIdx.x;
    const int lane = tid & 31;
    const int w    = tid >> 5;
    const float F1 = 0.5641895835477563f;  // 1/sqrt(pi)
    const float F2 = 1.1283791670955126f;  // 2/sqrt(pi)

    // ---- load per-atom data ----
    if (tid < kNA) {
        int g = mol * kNA + tid;
        int z = Z[g];
        sAlf[tid] = alpha[z];
        sJa[tid]  = Jp[z];
        sXi[tid]  = Xia[g];
        sRx[tid]  = R[3*g + 0];
        sRy[tid]  = R[3*g + 1];
        sRz[tid]  = R[3*g + 2];
    }
    __syncthreads();

    const float qtot = Qtot[mol];

    // ---- build augmented system [A | rhs] in LDS ----
    for (int idx = tid; idx < kNAug * (kNAug + 1); idx += kThreads) {
        int i = idx / (kNAug + 1);
        int j = idx - i * (kNAug + 1);
        float v;
        if (j == kNAug) {                       // rhs column
            v = (i < kNA) ? -sXi[i] : qtot;
        } else if (i == kNA && j == kNA) {
            v = 0.f;
        } else if (i == kNA || j == kNA) {
            v = 1.f;
        } else if (i == j) {
            v = sJa[i] + F2 * rsqrtf(2.f * sAlf[i] * sAlf[i]);
        } else {
            float dx = sRx[i]-sRx[j], dy = sRy[i]-sRy[j], dz = sRz[i]-sRz[j];
            float rij = sqrtf(fmaxf(dx*dx + dy*dy + dz*dz, 1e-12f));
            float g   = rsqrtf(sAlf[i]*sAlf[i] + sAlf[j]*sAlf[j]);
            v = erff(g * rij) / rij;
        }
        sA[i*kS + j] = v;
    }
    __syncthreads();

    // ---- LU forward elimination with partial pivoting (rhs carried as col 129) ----
    // Per column: wave 0 finds the pivot with shfl reductions and swaps rows
    // (no block barriers inside), then all 256 threads do a 2D-tiled rank-1
    // trailing update. 2 block barriers per column total.
    for (int k = 0; k < kNAug; ++k) {
        if (w == 0) {
            float bv = -1.f; int bi = k;
            for (int i = k + lane; i < kNAug; i += 32) {
                float v = fabsf(sA[i*kS + k]);
                if (v > bv) { bv = v; bi = i; }
            }
            for (int off = 16; off > 0; off >>= 1) {
                float ov = __shfl_down(bv, off, 32);
                int   oi = __shfl_down(bi, off, 32);
                if (ov > bv) { bv = ov; bi = oi; }
            }
            int p = __shfl(bi, 0, 32);
            if (p != k) {
                for (int j = k + lane; j <= kNAug; j += 32) {
                    float t0 = sA[k*kS + j];
                    sA[k*kS + j] = sA[p*kS + j];
                    sA[p*kS + j] = t0;
                }
            }
        }
        __syncthreads();

        float invP = 1.f / sA[k*kS + k];
        int nr = kNAug - 1 - k;            // rows k+1..128
        int nc = kNAug - k;                // cols k+1..129 (incl rhs)
        int tot = nr * nc;
        for (int t = tid; t < tot; t += kThreads) {
            int ri = t / nc;
            int cj = t - ri * nc;
            int i = k + 1 + ri;
            int j = k + 1 + cj;
            sA[i*kS + j] -= (sA[i*kS + k] * invP) * sA[k*kS + j];
        }
        __syncthreads();
    }

    // ---- back substitution (wave 0 only; shfl reduction per row) ----
    if (w == 0) {
        for (int k = kNAug - 1; k >= 0; --k) {
            float s = 0.f;
            for (int j = k + 1 + lane; j < kNAug; j += 32)
                s += sA[k*kS + j] * sQ[j];
            for (int off = 16; off > 0; off >>= 1)
                s += __shfl_down(s, off, 32);
            if (lane == 0)
                sQ[k] = (sA[k*kS + kNAug] - s) / sA[k*kS + k];
        }
    }
    __syncthreads();

    // ---- rebuild Coulomb kernel K (zero diagonal) into sA ----
    for (int idx = tid; idx < kNA * kNA; idx += kThreads) {
        int i = idx >> 7, j = idx & (kNA - 1);
        float kij = 0.f;
        if (i != j) {
            float dx = sRx[i]-sRx[j], dy = sRy[i]-sRy[j], dz = sRz[i]-sRz[j];
            float rij = sqrtf(fmaxf(dx*dx + dy*dy + dz*dz, 1e-12f));
            float g   = rsqrtf(sAlf[i]*sAlf[i] + sAlf[j]*sAlf[j]);
            kij = erff(g * rij) / rij;
        }
        sA[i*kS + j] = kij;
    }
    __syncthreads();

    // ---- y = K q via V_WMMA_F32_16X16X4_F32 (wave w owns row tile w) ----
    // A 16x4 layout: lane half h holds K = 2h + vgpr; M = lane&15.
    // B 4x16 (all columns equal = q chunk): VGPR v, half h -> K = 2h + v.
    // C/D 16x16: VGPR v, half h -> row = v + 8h, col = lane&15.
    {
        int h = lane >> 4;
        int m = (w << 4) + (lane & 15);
        v8f acc = {};
        for (int kc = 0; kc < kNA / 4; ++kc) {
            int kb = (kc << 2) + (h << 1);                 // even -> 8B aligned
            v2f a = *(const v2f*)(sA + m * kS + kb);
            v2f b = *(const v2f*)(sQ + kb);
            acc = __builtin_amdgcn_wmma_f32_16x16x4_f32(
                false, a, false, b, (short)0, acc, false, false);
        }
        if ((lane & 15) == 0) {
            int base = (w << 4) + (h << 3);
            #pragma unroll
            for (int v = 0; v < 8; ++v) sY[base + v] = acc[v];
        }
    }
    __syncthreads();

    // ---- per-atom outputs: Qa then Ea (concatenated) ----
    if (tid < kNA) {
        int g = mol * kNA + tid;
        float q   = sQ[tid];
        float gii = rsqrtf(2.f * sAlf[tid] * sAlf[tid]);
        float ea  = sXi[tid]*q + (0.5f*sJa[tid] + F1*gii)*q*q + 0.5f*q*sY[tid];
        out[g]     = q;
        out[T + g] = ea;
    }
}

extern "C" void kernel_launch(void* const* d_in, const int* in_sizes, int n_in,
                              void* d_out, int out_size, void* d_ws, size_t ws_size,
                              hipStream_t stream) {
    // setup_inputs order: Z, R, Dij, idx_i, idx_j, Xia, Q_tot, batch_seg, alpha, Jp
    const int*   Z     = (const int*)  d_in[0];
    const float* R     = (const float*)d_in[1];
    const float* Xia   = (const float*)d_in[5];
    const float* Qtot  = (const float*)d_in[6];
    const float* alpha = (const float*)d_in[8];
    const float* Jp    = (const float*)d_in[9];
    float* out = (float*)d_out;

    const int T = in_sizes[0];   // total atoms
    const int M = in_sizes[6];   // molecules
    const size_t shmem = (size_t)kSmemFloats * sizeof(float);  // ~72 KB dynamic LDS

    eem_mol_kernel<<<M, kThreads, shmem, stream>>>(Z, R, Xia, Qtot, alpha, Jp, out, T);
}